// YOLOLoss_33225867002518
// MI455X (gfx1250) — compile-verified
//
#include <hip/hip_runtime.h>
#include <math.h>

// ---- constants from the reference ----
#define GH 5
#define GW 10
#define NC 80
#define NA 3
#define CH_PER_A 85                 // 5 + NUM_CLASSES
#define PLANE (GH*GW)               // 50 floats per channel plane
#define BSTRIDE (NA*CH_PER_A*PLANE) // 12750 floats per batch
#define IOU_TH 0.3f

typedef float v2f __attribute__((ext_vector_type(2)));
typedef float v4f __attribute__((ext_vector_type(4)));
typedef float v8f __attribute__((ext_vector_type(8)));

__device__ __forceinline__ float sigm(float x) { return 1.0f / (1.0f + expf(-x)); }
__device__ __forceinline__ float softplusf(float x) {
    // stable: max(x,0) + log1p(exp(-|x|))
    return fmaxf(x, 0.0f) + log1pf(expf(-fabsf(x)));
}
__device__ __forceinline__ float wred_sum(float v) {
    #pragma unroll
    for (int m = 16; m > 0; m >>= 1) v += __shfl_xor(v, m, 32);
    return v;
}
__device__ __forceinline__ float wred_max(float v) {
    #pragma unroll
    for (int m = 16; m > 0; m >>= 1) v = fmaxf(v, __shfl_xor(v, m, 32));
    return v;
}

// One 256-thread block (8 wave32s) per batch. Wave t handles target t.
// Writes 4 partial sums per batch to ws: {coord, obj, noobj_inner, cls}.
__global__ void __launch_bounds__(256)
yolo_partial(const float* __restrict__ preds,
             const float* __restrict__ tgts,
             const float* __restrict__ anch,
             float* __restrict__ part,
             int B, int T) {
    const int b    = blockIdx.x;
    const int tid  = threadIdx.x;
    const int wave = tid >> 5;
    const int lane = tid & 31;

    __shared__ float s_wred[8];
    __shared__ float s_S;
    __shared__ float s_out[8][4];

    const float* __restrict__ pb = preds + (size_t)b * BSTRIDE;

    // ---- Phase 1: S = sum over (a,y,x) of softplus(sigmoid(pred_obj)) ----
    float sp = 0.0f;
    if (tid < NA * PLANE) {                      // 150 elements
        int a = tid / PLANE, r = tid - a * PLANE;
        float v = pb[(a * CH_PER_A + 4) * PLANE + r];
        sp = softplusf(sigm(v));
    }
    float wsum = wred_sum(sp);
    if (lane == 0) s_wred[wave] = wsum;
    __syncthreads();
    if (tid == 0) {
        float s = 0.0f;
        #pragma unroll
        for (int i = 0; i < 8; ++i) s += s_wred[i];
        s_S = s;
    }
    __syncthreads();
    const float S = s_S;

    // ---- Phase 2: per-target terms (wave-parallel over classes) ----
    float acc0 = 0.0f, acc1 = 0.0f, acc2 = 0.0f, acc3 = 0.0f;
    for (int t = wave; t < T; t += 8) {
        const float* tg = tgts + ((size_t)b * T + t) * 5;
        const int   cls = (int)tg[0];
        const float tx = tg[1], ty = tg[2], tw = tg[3], th = tg[4];

        // best anchor by IOU (first max wins, like argmax)
        int best_a = 0; float best_iou = -1.0f;
        #pragma unroll
        for (int a = 0; a < NA; ++a) {
            float aw = anch[2*a], ah = anch[2*a+1];
            float inter = fminf(tw, aw) * fminf(th, ah);
            float uni   = tw * th + aw * ah - inter;
            float iou   = inter / (uni + 1e-6f);
            if (iou > best_iou) { best_iou = iou; best_a = a; }
        }
        int gx = (int)floorf(tx * (float)GW);
        int gy = (int)floorf(ty * (float)GH);
        const float vm = ((gx < GW) && (gy < GH) && (best_iou >= IOU_TH)) ? 1.0f : 0.0f;
        int gxc = min(max(gx, 0), GW - 1);
        int gyc = min(max(gy, 0), GH - 1);

        // base of the matched cell; channel c lives at pcell[c*PLANE]
        const float* __restrict__ pcell =
            pb + best_a * (CH_PER_A * PLANE) + gyc * GW + gxc;

        // class logits: lane handles classes {lane, lane+32, lane+64(<80)}
        const bool has2 = (lane < NC - 64);      // lane < 16
        float l0 = pcell[(5 + lane) * PLANE];
        float l1 = pcell[(5 + lane + 32) * PLANE];
        float l2 = has2 ? pcell[(5 + lane + 64) * PLANE] : -3.4e38f;
        float M  = wred_max(fmaxf(l0, fmaxf(l1, l2)));
        float se = expf(l0 - M) + expf(l1 - M) + (has2 ? expf(l2 - M) : 0.0f);
        float tsel = 0.0f;
        if      (cls == lane)              tsel = l0;
        else if (cls == lane + 32)         tsel = l1;
        else if (has2 && cls == lane + 64) tsel = l2;
        se   = wred_sum(se);
        tsel = wred_sum(tsel);
        const float ce = M + logf(se) - tsel;

        // box head: lanes 0..4 gather p[0..4]
        float pv = (lane < 5) ? pcell[lane * PLANE] : 0.0f;
        const float aw_s = anch[2 * best_a], ah_s = anch[2 * best_a + 1];
        float tgtv = 0.0f;
        if      (lane == 0) tgtv = tx * (float)GW - (float)gx;
        else if (lane == 1) tgtv = ty * (float)GH - (float)gy;
        else if (lane == 2) tgtv = logf(tw / aw_s + 1e-6f);
        else if (lane == 3) tgtv = logf(th / ah_s + 1e-6f);
        float d  = pv - tgtv;
        float coord = wred_sum((lane < 4) ? d * d : 0.0f);

        float p4  = __shfl(pv, 4, 32);
        float z   = sigm(p4);
        float objt = softplusf(-z);
        float spm  = softplusf(z);

        acc0 += vm * coord;
        acc1 += vm * objt;
        acc2 += vm * (S - spm);
        acc3 += vm * ce;
    }
    if (lane == 0) {
        s_out[wave][0] = acc0; s_out[wave][1] = acc1;
        s_out[wave][2] = acc2; s_out[wave][3] = acc3;
    }
    __syncthreads();
    if (tid == 0) {
        float c0 = 0, c1 = 0, c2 = 0, c3 = 0;
        #pragma unroll
        for (int w = 0; w < 8; ++w) {
            c0 += s_out[w][0]; c1 += s_out[w][1];
            c2 += s_out[w][2]; c3 += s_out[w][3];
        }
        float* o = part + (size_t)b * 4;
        o[0] = c0; o[1] = c1; o[2] = c2; o[3] = c3;
    }
}

// Single wave32. Deterministic reduction of B x 4 partials via
// V_WMMA_F32_16X16X4_F32 row-sum accumulation: D += A(16x4) x ones(4x16).
// A layout (ISA 7.12.2): lanes 0-15 -> M=lane, K=0/1; lanes 16-31 -> M=lane-16, K=2/3.
// Each lane owns two whole batch rows (contiguous float4s -> two b128 loads),
// bulk loop is unguarded; only the (rare) tail chunk carries bounds selects.
__global__ void __launch_bounds__(32)
yolo_finalize(const float* __restrict__ part, float* __restrict__ out, int B, int T) {
    const int lane = threadIdx.x;
    const int m    = lane & 15;
    const int koff = (lane < 16) ? 0 : 2;

    v8f acc[4];
    #pragma unroll
    for (int c = 0; c < 4; ++c) acc[c] = (v8f){0,0,0,0,0,0,0,0};
    v2f ones; ones.x = 1.0f; ones.y = 1.0f;

    const int Bfull = (B >> 6) << 6;           // full 64-batch chunks
    for (int cb = 0; cb < Bfull; cb += 64) {
        const int r0 = cb + m * 4 + koff;
        const v4f q0 = *(const v4f*)(part + (size_t)r0 * 4);       // batch r0, 4 comps
        const v4f q1 = *(const v4f*)(part + (size_t)(r0 + 1) * 4); // batch r0+1
        #pragma unroll
        for (int c = 0; c < 4; ++c) {
            v2f a; a.x = q0[c]; a.y = q1[c];
            acc[c] = __builtin_amdgcn_wmma_f32_16x16x4_f32(
                false, a, false, ones, (short)0, acc[c], false, false);
        }
    }
    if (Bfull < B) {                            // single guarded tail chunk
        const int r0 = Bfull + m * 4 + koff;
        const int r1 = r0 + 1;
        #pragma unroll
        for (int c = 0; c < 4; ++c) {
            v2f a;
            a.x = (r0 < B) ? part[(size_t)r0 * 4 + c] : 0.0f;
            a.y = (r1 < B) ? part[(size_t)r1 * 4 + c] : 0.0f;
            acc[c] = __builtin_amdgcn_wmma_f32_16x16x4_f32(
                false, a, false, ones, (short)0, acc[c], false, false);
        }
    }

    // D[m,n] is the row-sum replicated across n; fold 8 VGPRs then swap halves.
    float sums[4];
    #pragma unroll
    for (int c = 0; c < 4; ++c) {
        float p = 0.0f;
        #pragma unroll
        for (int r = 0; r < 8; ++r) p += acc[c][r];
        sums[c] = p + __shfl_xor(p, 16, 32);
    }

    if (lane == 0) {
        const float total_boxes = (float)B * (float)T;
        const float loss_coord = sums[0] / total_boxes;
        const float loss_obj   = sums[1];
        const float loss_noobj = 0.5f * sums[2];          // LAMBDA_NOOBJ inside
        const float loss_cls   = sums[3] / total_boxes;
        const float total = 5.0f * loss_coord + 1.0f * loss_obj +
                            0.5f * loss_noobj + 1.0f * loss_cls;
        out[0] = total;
        out[1] = loss_coord;
        out[2] = loss_obj;
        out[3] = loss_noobj;
        out[4] = loss_cls;
    }
}

extern "C" void kernel_launch(void* const* d_in, const int* in_sizes, int n_in,
                              void* d_out, int out_size, void* d_ws, size_t ws_size,
                              hipStream_t stream) {
    const float* preds = (const float*)d_in[0];
    const float* tgts  = (const float*)d_in[1];
    const float* anch  = (const float*)d_in[2];
    float* part = (float*)d_ws;
    float* out  = (float*)d_out;

    const int B = in_sizes[0] / BSTRIDE;          // 4096
    const int T = in_sizes[1] / (B * 5);          // 8

    yolo_partial<<<B, 256, 0, stream>>>(preds, tgts, anch, part, B, T);
    yolo_finalize<<<1, 32, 0, stream>>>(part, out, B, T);
}